// GraphAug_48541720379667
// MI455X (gfx1250) — compile-verified
//
#include <hip/hip_runtime.h>
#include <stdint.h>

// ---------------- problem constants (from reference) ----------------
#define N_USER_   200000
#define N_ITEM_   100000
#define N_NODES_  300000
#define DIM_      64
#define NNZ_      6000000
#define N_LAYERS_ 3
#define SLOPE_    0.5f

#define TILE        256     // edges staged per block per iteration
#define SPMM_BLOCKS 4096

// ---------------- CDNA5 async global->LDS path (guarded) ----------------
#if defined(__HIP_DEVICE_COMPILE__) && __has_builtin(__builtin_amdgcn_global_load_async_to_lds_b32)
#define HAVE_ASYNC_LDS 1
#else
#define HAVE_ASYNC_LDS 0
#endif

#if defined(__HIP_DEVICE_COMPILE__) && __has_builtin(__builtin_amdgcn_s_wait_asynccnt)
#define WAIT_ASYNC(n) do { __builtin_amdgcn_s_wait_asynccnt(n); asm volatile("" ::: "memory"); } while (0)
#elif defined(__HIP_DEVICE_COMPILE__)
#define WAIT_ASYNC(n) asm volatile("s_wait_asynccnt %0" :: "i"(n) : "memory")
#else
#define WAIT_ASYNC(n) do {} while (0)
#endif

typedef __attribute__((address_space(1))) int g1_i32;   // global (AS1) int
typedef __attribute__((address_space(3))) int l3_i32;   // LDS (AS3) int

__device__ __forceinline__ void async_g2l_b32(const uint32_t* gp, uint32_t* lp) {
#if HAVE_ASYNC_LDS
    // per-lane 4B async copy global -> LDS, tracked by ASYNCcnt
    __builtin_amdgcn_global_load_async_to_lds_b32((g1_i32*)gp, (l3_i32*)lp, 0, 0);
#else
    *lp = *gp;   // fallback: plain copy (compiler tracks DScnt dependency)
#endif
}

__device__ __forceinline__ void atomic_fadd(float* p, float v) {
#if defined(__HIP_DEVICE_COMPILE__)
    unsafeAtomicAdd(p, v);   // native global_atomic_add_f32
#else
    atomicAdd(p, v);
#endif
}

// ---------------- lowering probe (never launched; first in file so the
// --disasm snippet shows the s_wait_asynccnt and f32-atomic lowerings) -------
__global__ void aaa_probe_kernel(float* p) {
    WAIT_ASYNC(0);
    atomic_fadd(p + threadIdx.x, 1.0f);
}

// ---------------- kernels ----------------

// y[r] += v * x[c] over all edges.  One wave per edge; lane l covers dims 2l,2l+1.
// Edge metadata staged into LDS via double-buffered async copies; each wave
// consumes exactly the 32 LDS slots it issued loads for, so synchronization is
// a per-wave s_wait_asynccnt only (no barriers).
__global__ __launch_bounds__(256) void spmm_kernel(const long long* __restrict__ adj_index, // [2, NNZ]
                                                   const float*     __restrict__ vals,
                                                   const float*     __restrict__ x,
                                                   float*           __restrict__ y) {
    __shared__ uint32_t s_row[2][TILE];
    __shared__ uint32_t s_col[2][TILE];
    __shared__ float    s_val[2][TILE];

    // int64 indices are < 300000: read only the low dword of each element
    const uint32_t* rows_lo = (const uint32_t*)adj_index;           // row e @ dword 2e
    const uint32_t* cols_lo = (const uint32_t*)(adj_index + NNZ_);  // col e @ dword 2e
    const uint32_t* vals_u  = (const uint32_t*)vals;

    const int tid  = (int)threadIdx.x;
    const int wave = tid >> 5;
    const int lane = tid & 31;
    const int ntiles = (NNZ_ + TILE - 1) / TILE;

    auto issue = [&](int t, int buf) {
        int e = t * TILE + tid;
        if (e > NNZ_ - 1) e = NNZ_ - 1;          // clamp: keep EXEC full so each
        const size_t de = 2 * (size_t)e;          // wave always issues 3 async ops
        async_g2l_b32(rows_lo + de, &s_row[buf][tid]);
        async_g2l_b32(cols_lo + de, &s_col[buf][tid]);
        async_g2l_b32(vals_u + e, (uint32_t*)&s_val[buf][tid]);
    };

    int t = (int)blockIdx.x;
    if (t >= ntiles) return;
    int buf = 0;
    issue(t, 0);

    for (; t < ntiles; t += (int)gridDim.x, buf ^= 1) {
        const int tn   = t + (int)gridDim.x;
        const bool more = (tn < ntiles);
        if (more) {
            issue(tn, buf ^ 1);   // prefetch next tile into the other buffer
            WAIT_ASYNC(3);        // 3 newer (next-tile) ops may stay in flight
        } else {
            WAIT_ASYNC(0);
        }

        const int base = wave * 32;   // this wave's own slots
        #pragma unroll 4
        for (int k = 0; k < 32; ++k) {
            const int slot = base + k;
            const uint32_t r = s_row[buf][slot];
            const uint32_t c = s_col[buf][slot];
            float v          = s_val[buf][slot];
            if (t * TILE + slot >= NNZ_) v = 0.0f;   // tail edges add 0 to a valid row

            const float2 xv = *(const float2*)(x + (size_t)c * DIM_ + 2 * lane); // coalesced 256B gather
            float s0 = v * xv.x;
            float s1 = v * xv.y;
            float* yp = y + (size_t)r * DIM_ + 2 * lane;
            atomic_fadd(yp,     s0);
            atomic_fadd(yp + 1, s1);
        }
    }
}

// out = x = concat(uEmbeds, iEmbeds)   (float4 / b128 path)
__global__ __launch_bounds__(256) void init_kernel(const float4* __restrict__ u4,
                                                   const float4* __restrict__ i4,
                                                   float4* __restrict__ x4,
                                                   float4* __restrict__ out4) {
    const int nu4 = N_USER_ * DIM_ / 4;
    const int n4  = N_NODES_ * DIM_ / 4;
    int i = blockIdx.x * 256 + threadIdx.x;
    if (i < n4) {
        float4 v = (i < nu4) ? u4[i] : i4[i - nu4];
        x4[i]   = v;
        out4[i] = v;
    }
}

__global__ __launch_bounds__(256) void zero_kernel(float4* __restrict__ y4) {
    const int n4 = N_NODES_ * DIM_ / 4;
    int i = blockIdx.x * 256 + threadIdx.x;
    if (i < n4) y4[i] = make_float4(0.f, 0.f, 0.f, 0.f);
}

// x = leaky(y); out += x
__global__ __launch_bounds__(256) void combine_kernel(const float4* __restrict__ y4,
                                                      float4* __restrict__ x4,
                                                      float4* __restrict__ out4) {
    const int n4 = N_NODES_ * DIM_ / 4;
    int i = blockIdx.x * 256 + threadIdx.x;
    if (i < n4) {
        float4 v = y4[i];
        float4 l;
        l.x = (v.x >= 0.f) ? v.x : SLOPE_ * v.x;
        l.y = (v.y >= 0.f) ? v.y : SLOPE_ * v.y;
        l.z = (v.z >= 0.f) ? v.z : SLOPE_ * v.z;
        l.w = (v.w >= 0.f) ? v.w : SLOPE_ * v.w;
        x4[i] = l;
        float4 o = out4[i];
        o.x += l.x; o.y += l.y; o.z += l.z; o.w += l.w;
        out4[i] = o;
    }
}

// ---------------- launch ----------------
extern "C" void kernel_launch(void* const* d_in, const int* in_sizes, int n_in,
                              void* d_out, int out_size, void* d_ws, size_t ws_size,
                              hipStream_t stream) {
    (void)in_sizes; (void)n_in; (void)out_size; (void)ws_size;

    const long long* adj_index  = (const long long*)d_in[0]; // [2, NNZ] int64
    const float*     adj_values = (const float*)d_in[1];     // [NNZ]
    const float*     uE         = (const float*)d_in[2];     // [N_USER, 64]
    const float*     iE         = (const float*)d_in[3];     // [N_ITEM, 64]
    float*           out        = (float*)d_out;             // [N_NODES, 64]

    // workspace: x buffer + y buffer, 76.8 MB each
    float* xbuf = (float*)d_ws;
    float* ybuf = xbuf + (size_t)N_NODES_ * DIM_;

    const int n4 = N_NODES_ * DIM_ / 4;
    const int eb = (n4 + 255) / 256;

    init_kernel<<<eb, 256, 0, stream>>>((const float4*)uE, (const float4*)iE,
                                        (float4*)xbuf, (float4*)out);

    const int ntiles = (NNZ_ + TILE - 1) / TILE;
    const int sb = (ntiles < SPMM_BLOCKS) ? ntiles : SPMM_BLOCKS;

    for (int layer = 0; layer < N_LAYERS_; ++layer) {
        zero_kernel<<<eb, 256, 0, stream>>>((float4*)ybuf);
        spmm_kernel<<<sb, 256, 0, stream>>>(adj_index, adj_values, xbuf, ybuf);
        combine_kernel<<<eb, 256, 0, stream>>>((const float4*)ybuf,
                                               (float4*)xbuf, (float4*)out);
    }
}